// TokenKMerHead_63144609185804
// MI455X (gfx1250) — compile-verified
//
#include <hip/hip_runtime.h>

typedef __attribute__((ext_vector_type(2))) float v2f;
typedef __attribute__((ext_vector_type(8))) float v8f;

#define S_LEN 2048
#define H_DIM 1024
#define B_DIM 16
#define C_DIM 16
#define KMER  6
#define OUT_S (S_LEN + KMER - 1)   /* 2053 */

// ---------------------------------------------------------------------------
// Kernel 0: seq_len[b] = sum(attention_mask[b, :])   (jnp bool -> 1 byte/elem)
// ---------------------------------------------------------------------------
__global__ void kmer_seqlen(const unsigned char* __restrict__ mask,
                            int* __restrict__ sl) {
  __shared__ int red[256];
  const int b = blockIdx.x, t = threadIdx.x;
  int acc = 0;
  for (int i = t; i < S_LEN; i += 256) acc += (mask[b * S_LEN + i] != 0);
  red[t] = acc;
  __syncthreads();
  for (int off = 128; off > 0; off >>= 1) {
    if (t < off) red[t] += red[t + off];
    __syncthreads();
  }
  if (t == 0) sl[b] = red[0];
}

// ---------------------------------------------------------------------------
// Kernel 1: Y[b*S+s, c] = emb[b,s,:] . W[:,c]  via v_wmma_f32_16x16x4_f32
//   M = B*S = 32768, K = 1024, N = 16.
//   Each wave computes 2 M-tiles (32 rows) -> 1024 waves, spread over 256
//   blocks (4 waves/block) so every WGP hosts work. Unroll-8 K loop keeps
//   16 outstanding global_load_b64 per wave (~4 MB in flight device-wide)
//   to cover the HBM bandwidth-delay product at 23.3 TB/s.
//   A 16x4 f32 fragment: lane = M-row (0..15 per half-wave), VGPR v holds
//   K = 2*(lane>>4) + v.  B 4x16 fragment mirrors it with lane = N column.
// ---------------------------------------------------------------------------
__global__ void __launch_bounds__(128) kmer_gemm(const float* __restrict__ emb,
                                                 const float* __restrict__ W,
                                                 float* __restrict__ Y) {
  // 64 KB LDS copy of W, stored [k][c]: lanes stride contiguous c -> no bank
  // conflicts (upper half-wave reads k+2 -> +32 banks, still disjoint).
  __shared__ float Ws[H_DIM][C_DIM];
  for (int idx = threadIdx.x; idx < H_DIM * C_DIM; idx += 128)
    Ws[idx >> 4][idx & 15] = W[idx];
  __syncthreads();

  const int lane    = threadIdx.x & 31;
  const int wave    = blockIdx.x * 4 + (threadIdx.x >> 5);
  const int rowBase = wave * 32;            // two 16-row tiles
  const int n       = lane & 15;            // A row within tile == B column
  const int kOff    = (lane >> 4) << 1;     // 0 or 2

  const float* a0p = emb + (size_t)(rowBase + n) * H_DIM + kOff;
  const float* a1p = emb + (size_t)(rowBase + 16 + n) * H_DIM + kOff;

  v8f acc0 = {};
  v8f acc1 = {};

#pragma unroll 8
  for (int kk = 0; kk < H_DIM; kk += 4) {
    v2f a0 = *(const v2f*)(a0p + kk);       // 8B aligned (kOff even, kk%4==0)
    v2f a1 = *(const v2f*)(a1p + kk);
    v2f bf;
    bf.x = Ws[kk + kOff + 0][n];
    bf.y = Ws[kk + kOff + 1][n];
    // (neg_a, A, neg_b, B, c_mod, C, reuse_a, reuse_b)
    acc0 = __builtin_amdgcn_wmma_f32_16x16x4_f32(false, a0, false, bf,
                                                 (short)0, acc0, false, false);
    acc1 = __builtin_amdgcn_wmma_f32_16x16x4_f32(false, a1, false, bf,
                                                 (short)0, acc1, false, false);
  }

  // D layout: VGPR v, lane l -> m = v + 8*(l>>4), n = l&15
  const int mh = (lane >> 4) << 3;
#pragma unroll
  for (int v = 0; v < 8; ++v) {
    const int m = v + mh;
    Y[(size_t)(rowBase + m) * C_DIM + n]      = acc0[v];
    Y[(size_t)(rowBase + 16 + m) * C_DIM + n] = acc1[v];
  }
}

// ---------------------------------------------------------------------------
// Kernel 2: assemble output rows from Y (all of Y sits in L2).
//   out[b,i,c] = tok_row(b,i) . W[:,c] + bias[c], computed as projected means.
// ---------------------------------------------------------------------------
__global__ void kmer_assemble(const float* __restrict__ Y,
                              const int* __restrict__ sl,
                              const float* __restrict__ bias,
                              float* __restrict__ out) {
  const int idx = blockIdx.x * blockDim.x + threadIdx.x;
  if (idx >= B_DIM * OUT_S * C_DIM) return;
  const int c = idx & 15;
  const int i = (idx >> 4) % OUT_S;
  const int b = idx / (OUT_S * C_DIM);

  const int s = sl[b];
  const int L = s - 2;

  float val;
  if (i == s + KMER - 2) {                       // eos override (set after means)
    val = Y[((size_t)b * S_LEN + (s - 1)) * C_DIM + c];
  } else if (i == 0) {                           // bos
    val = Y[((size_t)b * S_LEN) * C_DIM + c];
  } else if (i <= S_LEN + KMER - 3) {            // means region, i in [1, 2051]
    const int ii  = i - 1;
    const int lo  = ii - KMER + 1 > 0 ? ii - KMER + 1 : 0;
    const int hi  = (L < ii + 1) ? L : ii + 1;
    const int cnt = hi - lo;
    if (cnt > 0) {
      float sum = 0.0f;
      for (int j = lo; j < hi; ++j)              // inner row j -> emb row j+1
        sum += Y[((size_t)b * S_LEN + j + 1) * C_DIM + c];
      val = sum / (float)cnt;
    } else {
      val = 0.0f;
    }
  } else {                                       // i == 2052 and not eos
    val = 0.0f;
  }
  out[idx] = val + bias[c];
}

// ---------------------------------------------------------------------------
extern "C" void kernel_launch(void* const* d_in, const int* in_sizes, int n_in,
                              void* d_out, int out_size, void* d_ws, size_t ws_size,
                              hipStream_t stream) {
  (void)in_sizes; (void)n_in; (void)out_size; (void)ws_size;

  const float*         emb  = (const float*)d_in[0];
  const unsigned char* mask = (const unsigned char*)d_in[1];
  const float*         W    = (const float*)d_in[2];
  const float*         bias = (const float*)d_in[3];

  float* Y  = (float*)d_ws;                              // 2 MB
  int*   sl = (int*)((char*)d_ws +
                     (size_t)B_DIM * S_LEN * C_DIM * sizeof(float));
  float* out = (float*)d_out;

  kmer_seqlen<<<B_DIM, 256, 0, stream>>>(mask, sl);

  const int gemmBlocks = (B_DIM * S_LEN) / (32 * 4);     // 256 blocks, 4 waves
  kmer_gemm<<<gemmBlocks, 128, 0, stream>>>(emb, W, Y);

  const int total = B_DIM * OUT_S * C_DIM;               // 525568
  kmer_assemble<<<(total + 255) / 256, 256, 0, stream>>>(Y, sl, bias, out);
}